// CompGCNConv_84439057039588
// MI455X (gfx1250) — compile-verified
//
#include <hip/hip_runtime.h>
#include <hip/hip_bf16.h>

#define DD 128  // feature dim

typedef __attribute__((ext_vector_type(2))) float v2f;
typedef __attribute__((ext_vector_type(8))) float v8f;

// CDNA5 fp32 WMMA: D[16x16] = A[16x4] * B[4x16] + C, full fp32 precision.
static __device__ __forceinline__ v8f wmma_f32(v2f a, v2f b, v8f c) {
  return __builtin_amdgcn_wmma_f32_16x16x4_f32(
      /*neg_a=*/false, a, /*neg_b=*/false, b,
      /*c_mod=*/(short)0, c, /*reuse_a=*/false, /*reuse_b=*/false);
}

static __device__ __forceinline__ void atomic_add_f32(float* p, float v) {
  __hip_atomic_fetch_add(p, v, __ATOMIC_RELAXED, __HIP_MEMORY_SCOPE_AGENT);
}

// ---------------------------------------------------------------- zero scratch
__global__ __launch_bounds__(256) void zero_kernel(float* __restrict__ p, size_t n) {
  size_t i = (size_t)blockIdx.x * blockDim.x + threadIdx.x;
  size_t stride = (size_t)gridDim.x * blockDim.x;
  for (; i < n; i += stride) p[i] = 0.0f;
}

// ------------------------------------------------- edge scatter (1 wave/edge)
// agg_O[dst] += ent[src] - rel ;  agg_I[src] += ent[dst] - rel ;  deg counts
__global__ __launch_bounds__(256) void edge_scatter_kernel(
    const float* __restrict__ ent, const float* __restrict__ relE,
    const int* __restrict__ src, const int* __restrict__ dst,
    const int* __restrict__ etyp,
    float* __restrict__ aggO, float* __restrict__ aggI, float* __restrict__ deg,
    int E) {
  int tid = blockIdx.x * blockDim.x + threadIdx.x;
  int e = tid >> 5;
  int lane = tid & 31;
  if (e >= E) return;
  int s = src[e], d = dst[e], t = etyp[e];

  float4 r4 = ((const float4*)(relE + (size_t)t * DD))[lane];
  float4 es = ((const float4*)(ent + (size_t)s * DD))[lane];
  float4 ed = ((const float4*)(ent + (size_t)d * DD))[lane];

  float* po = aggO + (size_t)d * DD + lane * 4;
  atomic_add_f32(po + 0, es.x - r4.x);
  atomic_add_f32(po + 1, es.y - r4.y);
  atomic_add_f32(po + 2, es.z - r4.z);
  atomic_add_f32(po + 3, es.w - r4.w);

  float* pi = aggI + (size_t)s * DD + lane * 4;
  atomic_add_f32(pi + 0, ed.x - r4.x);
  atomic_add_f32(pi + 1, ed.y - r4.y);
  atomic_add_f32(pi + 2, ed.z - r4.z);
  atomic_add_f32(pi + 3, ed.w - r4.w);

  if (lane == 0) {
    atomic_add_f32(&deg[d], 1.0f);
    atomic_add_f32(&deg[s], 1.0f);
  }
}

// ---------------------------------------------------------- fused node GEMM
// out_pre[n,:] = (X@WS^T + AO@WO^T + AI@WI^T)[n,:] / max(deg[n],1)
// One wave -> one 16x16 output tile; 8 waves/block cover all 128 columns.
__global__ __launch_bounds__(256) void fused_gemm_kernel(
    const float* __restrict__ X, const float* __restrict__ AO,
    const float* __restrict__ AI,
    const float* __restrict__ WS, const float* __restrict__ WO,
    const float* __restrict__ WI,
    const float* __restrict__ deg, float* __restrict__ out_pre) {
  const int lane = threadIdx.x & 31;
  const int wave = threadIdx.x >> 5;      // 0..7 -> column tile
  const int row0 = blockIdx.x * 16;
  const int n0 = wave * 16;
  const int m = lane & 15;
  const int sel = lane >> 4;              // lanes 16-31 hold K+2
  const int koff = sel * 2;

  const float* xr = X + (size_t)(row0 + m) * DD + koff;
  const float* aor = AO + (size_t)(row0 + m) * DD + koff;
  const float* air = AI + (size_t)(row0 + m) * DD + koff;
  const float* wsr = WS + (size_t)(n0 + m) * DD + koff;  // B[k][n]=W[n*128+k]
  const float* wor = WO + (size_t)(n0 + m) * DD + koff;
  const float* wir = WI + (size_t)(n0 + m) * DD + koff;

  v8f c0 = {}, c1 = {}, c2 = {};
#pragma unroll 4
  for (int kb = 0; kb < DD; kb += 4) {
    c0 = wmma_f32(*(const v2f*)(xr + kb),  *(const v2f*)(wsr + kb), c0);
    c1 = wmma_f32(*(const v2f*)(aor + kb), *(const v2f*)(wor + kb), c1);
    c2 = wmma_f32(*(const v2f*)(air + kb), *(const v2f*)(wir + kb), c2);
  }

  const int n = n0 + (lane & 15);
#pragma unroll
  for (int r = 0; r < 8; ++r) {
    int row = row0 + r + 8 * sel;   // D layout: VGPR r = M r (lanes 0-15) / r+8
    float dg = deg[row];
    dg = dg < 1.0f ? 1.0f : dg;
    out_pre[(size_t)row * DD + n] = (c0[r] + c1[r] + c2[r]) / dg;
  }
}

// ----------------------------------------------------- per-feature statistics
#define RPB 512
__global__ __launch_bounds__(128) void colstats_kernel(
    const float* __restrict__ out_pre, float* __restrict__ sums,
    float* __restrict__ ssqs, int nrows) {
  int d = threadIdx.x;                  // 128 threads = 128 features
  int r0 = blockIdx.x * RPB;
  int r1 = r0 + RPB; if (r1 > nrows) r1 = nrows;
  float s = 0.0f, q = 0.0f;
  for (int r = r0; r < r1; ++r) {
    float v = out_pre[(size_t)r * DD + d];   // coalesced across threads
    s += v; q += v * v;
  }
  atomic_add_f32(&sums[d], s);
  atomic_add_f32(&ssqs[d], q);
}

// ------------------------------------------------------------- BN + ReLU out
__global__ __launch_bounds__(256) void bn_relu_kernel(
    const float* __restrict__ out_pre, const float* __restrict__ sums,
    const float* __restrict__ ssqs, const float* __restrict__ gamma,
    const float* __restrict__ beta, float* __restrict__ out, int nrows) {
  size_t i = (size_t)blockIdx.x * blockDim.x + threadIdx.x;
  if (i >= (size_t)nrows * DD) return;
  int d = (int)(i & (DD - 1));
  float invn = 1.0f / (float)nrows;
  float mean = sums[d] * invn;
  float var = ssqs[d] * invn - mean * mean;
  float v = (out_pre[i] - mean) * rsqrtf(var + 1e-5f) * gamma[d] + beta[d];
  out[i] = v > 0.0f ? v : 0.0f;
}

// ------------------------------------------------ relation GEMM (R=200 rows)
__global__ __launch_bounds__(256) void rel_gemm_kernel(
    const float* __restrict__ R, const float* __restrict__ W,
    float* __restrict__ out, int nrows) {
  const int lane = threadIdx.x & 31;
  const int wave = threadIdx.x >> 5;
  const int row0 = blockIdx.x * 16;
  const int n0 = wave * 16;
  const int m = lane & 15;
  const int sel = lane >> 4;
  const int koff = sel * 2;

  int rload = row0 + m; if (rload > nrows - 1) rload = nrows - 1;  // clamp, keep EXEC full
  const float* rr = R + (size_t)rload * DD + koff;
  const float* wr = W + (size_t)(n0 + m) * DD + koff;

  v8f c = {};
#pragma unroll 4
  for (int kb = 0; kb < DD; kb += 4)
    c = wmma_f32(*(const v2f*)(rr + kb), *(const v2f*)(wr + kb), c);

  const int n = n0 + (lane & 15);
#pragma unroll
  for (int r = 0; r < 8; ++r) {
    int row = row0 + r + 8 * sel;
    if (row < nrows) out[(size_t)row * DD + n] = c[r];
  }
}

// ---------------------------------------------------------------- launch
extern "C" void kernel_launch(void* const* d_in, const int* in_sizes, int n_in,
                              void* d_out, int out_size, void* d_ws, size_t ws_size,
                              hipStream_t stream) {
  const float* ent   = (const float*)d_in[0];
  const float* relE  = (const float*)d_in[1];
  const int*   eidx  = (const int*)d_in[2];
  const int*   etyp  = (const int*)d_in[3];
  const float* WO    = (const float*)d_in[4];
  const float* WI    = (const float*)d_in[5];
  const float* WS    = (const float*)d_in[6];
  const float* Wrel  = (const float*)d_in[7];
  const float* gamma = (const float*)d_in[8];
  const float* beta  = (const float*)d_in[9];

  const int N = in_sizes[0] / DD;   // 50000
  const int R = in_sizes[1] / DD;   // 200
  const int E = in_sizes[3];        // 600000
  const int* src = eidx;            // edge_index row 0
  const int* dst = eidx + E;        // edge_index row 1

  float* ws = (float*)d_ws;
  const size_t ND = (size_t)N * DD;
  float* aggO = ws;                 // [N,128]
  float* aggI = ws + ND;            // [N,128]
  float* outp = ws + 2 * ND;        // [N,128] pre-BN
  float* deg  = ws + 3 * ND;        // [N]
  float* sums = deg + N;            // [128]
  float* ssqs = sums + DD;          // [128]

  float* out1 = (float*)d_out;      // [N,128]
  float* out2 = out1 + ND;          // [R,128]

  // re-zero accumulators every call (graph-replay safe)
  zero_kernel<<<2048, 256, 0, stream>>>(aggO, 2 * ND);
  zero_kernel<<<64, 256, 0, stream>>>(deg, (size_t)N + 2 * DD);

  int eth = E * 32;
  edge_scatter_kernel<<<(eth + 255) / 256, 256, 0, stream>>>(
      ent, relE, src, dst, etyp, aggO, aggI, deg, E);

  fused_gemm_kernel<<<N / 16, 256, 0, stream>>>(ent, aggO, aggI, WS, WO, WI,
                                                deg, outp);

  colstats_kernel<<<(N + RPB - 1) / RPB, 128, 0, stream>>>(outp, sums, ssqs, N);

  bn_relu_kernel<<<(int)((ND + 255) / 256), 256, 0, stream>>>(
      outp, sums, ssqs, gamma, beta, out1, N);

  rel_gemm_kernel<<<(R + 15) / 16, 256, 0, stream>>>(relE, Wrel, out2, R);
}